// FilterModel_25237227831811
// MI455X (gfx1250) — compile-verified
//
#include <hip/hip_runtime.h>
#include <stdint.h>

// Problem constants from the reference: B, N, C = (32, 8192, 512), K = 128.
#define B_ 32
#define N_ 8192
#define C_ 512
#define K_ 128
#define NTHREADS 1024
#define NWAVES (NTHREADS / 32)
#define NCHUNKS (N_ / NTHREADS)

typedef unsigned int u32x4 __attribute__((ext_vector_type(4)));
typedef int          i32x8 __attribute__((ext_vector_type(8)));
typedef int          i32x4 __attribute__((ext_vector_type(4)));

// One workgroup per batch row b:
//  1) TDM-gather column `id` of one_hot[b] (stride C floats) into LDS: tile is
//     1 element wide x N rows, tensor_dim0_stride = C  -> tensor_load_to_lds.
//  2) Coalesced store of the column as block_id output.
//  3) Stable stream compaction of non-zero positions -> indices output
//     (stored as float values: harness materializes all non-bf16 outputs f32).
__global__ __launch_bounds__(NTHREADS) void filter_gather_compact_kernel(
    const float* __restrict__ one_hot,
    const int*   __restrict__ id_p,
    float*       __restrict__ block_id_out,  // B*N floats
    float*       __restrict__ idx_out)       // B*K index values (as f32)
{
  __shared__ float    col[N_];          // 32 KB of the WGP's 320 KB LDS
  __shared__ unsigned wave_cnt[NWAVES];
  __shared__ unsigned base;

  const int b    = blockIdx.x;
  const int tid  = threadIdx.x;
  const int wid  = tid >> 5;
  const int lane = tid & 31;

  // ---- Wave 0 issues the Tensor Data Mover descriptor ----------------------
  if (tid < 32) {
    const int id = *id_p;
    const uint64_t gaddr =
        (uint64_t)(uintptr_t)(one_hot + ((size_t)b * N_ * C_ + (size_t)id));
    const unsigned lds_addr = (unsigned)(uintptr_t)(void*)&col[0];

    // D# Group 0 (128b): [1:0] count=1 | [63:32] lds_addr |
    //                    [120:64] global_addr | [127:126] type=2
    u32x4 g0;
    g0.x = 1u;
    g0.y = lds_addr;
    g0.z = (unsigned)(gaddr & 0xFFFFFFFFu);
    g0.w = ((unsigned)(gaddr >> 32) & 0x01FFFFFFu) | (2u << 30);

    // D# Group 1 (256b):
    //  [15:0] wg_mask=0 (not in cluster) | [17:16] data_size=2 (4B)
    //  [79:48] tensor_dim0=C | [111:80] tensor_dim1=N
    //  [127:112] tile_dim0=1 | [143:128] tile_dim1=N | [159:144] tile_dim2=0
    //  [207:160] tensor_dim0_stride=C | rest 0
    i32x8 g1;
    g1[0] = (int)(2u << 16);
    g1[1] = (int)(((unsigned)C_ & 0xFFFFu) << 16);
    g1[2] = (int)(((unsigned)C_ >> 16) | (((unsigned)N_ & 0xFFFFu) << 16));
    g1[3] = (int)(((unsigned)N_ >> 16) | (1u << 16));
    g1[4] = (int)((unsigned)N_ & 0xFFFFu);
    g1[5] = (int)(unsigned)C_;
    g1[6] = 0;
    g1[7] = 0;

    i32x4 z4 = {0, 0, 0, 0};                     // groups 2/3 unused (2D tensor)
    i32x8 z8 = {0, 0, 0, 0, 0, 0, 0, 0};         // extra group (clang-23 form)

    __builtin_amdgcn_tensor_load_to_lds(g0, g1, z4, z4, z8, /*cpol=*/0);
    __builtin_amdgcn_s_wait_tensorcnt(0);

    if (tid == 0) base = 0u;
  }
  __syncthreads();
  // The TDM wrote `col` behind the compiler's back; force it to treat LDS as
  // clobbered so the loads below are not optimized against "never written".
  asm volatile("" ::: "memory");

  const size_t out_base = (size_t)b * N_;

  // ---- Emit block_id + stable compaction, chunk by chunk (order-preserving)
  for (int c = 0; c < NCHUNKS; ++c) {
    const int   pos = c * NTHREADS + tid;
    const float v   = col[pos];

    block_id_out[out_base + pos] = v;  // coalesced 4B stores

    const bool     p    = (v != 0.0f);
    const unsigned mask = __builtin_amdgcn_ballot_w32(p);
    if (lane == 0) wave_cnt[wid] = (unsigned)__popc(mask);
    __syncthreads();

    unsigned prefix = base;
    for (int w = 0; w < wid; ++w) prefix += wave_cnt[w];
    const unsigned rank = (unsigned)__popc(mask & ((1u << lane) - 1u));

    if (p) {
      const unsigned idx = prefix + rank;
      if (idx < K_) idx_out[(size_t)b * K_ + idx] = (float)pos;
    }
    __syncthreads();

    if (tid == 0) {
      unsigned tot = 0;
      for (int w = 0; w < NWAVES; ++w) tot += wave_cnt[w];
      base += tot;
    }
    __syncthreads();
  }
}

extern "C" void kernel_launch(void* const* d_in, const int* in_sizes, int n_in,
                              void* d_out, int out_size, void* d_ws, size_t ws_size,
                              hipStream_t stream) {
  (void)in_sizes; (void)n_in; (void)out_size; (void)d_ws; (void)ws_size;

  const float* one_hot = (const float*)d_in[0];
  const int*   id_p    = (const int*)d_in[1];

  // Tuple output, concatenated flat: block_id (B*N) then indices (B*K).
  float* block_id_out = (float*)d_out;
  float* idx_out      = (float*)d_out + (size_t)B_ * N_;

  hipLaunchKernelGGL(filter_gather_compact_kernel, dim3(B_), dim3(NTHREADS), 0,
                     stream, one_hot, id_p, block_id_out, idx_out);
}